// DecoupledGIN_38817914421908
// MI455X (gfx1250) — compile-verified
//
#include <hip/hip_runtime.h>
#include <math.h>

#define NN 100000     // nodes
#define NE 1600000    // edges
#define DIN 128
#define DH 64
#define DSE 16
#define DOUT 16
#define NL 2
#define NG 512

typedef float v2f __attribute__((ext_vector_type(2)));
typedef float v8f __attribute__((ext_vector_type(8)));

// ---------------------------------------------------------------------------
// fp32 WMMA GEMM:  C[M,NOUT] = act( (A (+A2 if FUSE))[M,K] @ W[K,NOUT] + bias )
//
// Block = 256 threads = 8 waves computing a 32x64 macro-tile:
//   wave w -> row-tile (w>>2), col-tile (w&3), each a 16x16 WMMA tile.
// Both the 32xK A-tile and the KxNOUT weight are staged in LDS (A-tile row
// stride padded by 4 floats -> conflict-free ds reads). Inner loop is pure
// DS loads + V_WMMA_F32_16X16X4_F32 (fp32-exact, matches reference numerics).
// Requires M % 32 == 0 (EXEC must stay all-ones for WMMA).
// ---------------------------------------------------------------------------
template <int K, int NOUT, int ACT, bool FUSE>  // ACT: 0=none, 1=relu
__global__ __launch_bounds__(256) void gemm_f32_wmma(
    const float* __restrict__ A, const float* __restrict__ A2,
    const float* __restrict__ W, const float* __restrict__ bias,
    float* __restrict__ C) {
  constexpr int AS = K + 4;  // padded LDS row stride (floats)
  __shared__ float wlds[K * NOUT];
  __shared__ float alds[32 * AS];

  const int tid     = threadIdx.x;
  const int rowbase = blockIdx.x * 32;

  // Stage weights (K x NOUT) into LDS.
  for (int i = tid; i < K * NOUT; i += 256) wlds[i] = W[i];

  // Stage 32 x K A-tile into LDS, coalesced float4, optional fused A2 add.
  constexpr int K4 = K / 4;
  for (int j = tid; j < 32 * K4; j += 256) {
    const int r = j / K4, c = j % K4;
    float4 v = ((const float4*)(A + (size_t)(rowbase + r) * K))[c];
    if (FUSE) {
      float4 u = ((const float4*)(A2 + (size_t)(rowbase + r) * K))[c];
      v.x += u.x; v.y += u.y; v.z += u.z; v.w += u.w;
    }
    *(float4*)&alds[r * AS + 4 * c] = v;
  }
  __syncthreads();

  const int lane  = tid & 31;
  const int wave  = tid >> 5;
  const int half  = lane >> 4;  // 0 or 1
  const int m     = lane & 15;  // A row within tile == B/C col within tile
  const int kh    = half * 2;   // lanes 16-31 carry K+2,K+3 of each 4-block
  const int rtile = wave >> 2;  // 0..1
  const int col0  = (wave & 3) * 16;

  const float* __restrict__ arow = &alds[(rtile * 16 + m) * AS];

  v8f acc = {0.f, 0.f, 0.f, 0.f, 0.f, 0.f, 0.f, 0.f};
#pragma unroll
  for (int k = 0; k < K; k += 4) {
    v2f a, b;
    a.x = arow[k + kh];
    a.y = arow[k + kh + 1];
    b.x = wlds[(k + kh) * NOUT + col0 + m];
    b.y = wlds[(k + kh + 1) * NOUT + col0 + m];
    acc = __builtin_amdgcn_wmma_f32_16x16x4_f32(false, a, false, b, (short)0,
                                                acc, false, false);
  }

  const int   ncol = col0 + m;
  const float bv   = bias ? bias[ncol] : 0.0f;
#pragma unroll
  for (int r = 0; r < 8; ++r) {
    int   row = rowbase + rtile * 16 + r + half * 8;
    float v   = acc[r] + bv;
    if (ACT == 1) v = fmaxf(v, 0.0f);
    C[(size_t)row * NOUT + ncol] = v;
  }
}

// ---------------------------------------------------------------------------
// Degree / normalization
// ---------------------------------------------------------------------------
__global__ void edge_deg_kernel(const int* __restrict__ ei,
                                float* __restrict__ deg) {
  int e = blockIdx.x * blockDim.x + threadIdx.x;
  if (e < NE) atomicAdd(&deg[ei[NE + e]], 1.0f);
}

__global__ void dinv_kernel(const float* __restrict__ deg,
                            float* __restrict__ dinv) {
  int n = blockIdx.x * blockDim.x + threadIdx.x;
  if (n < NN) dinv[n] = rsqrtf(deg[n] + 1.0f);
}

// ---------------------------------------------------------------------------
// concat(x,s) -> h[N,128] (float4 granularity)
// ---------------------------------------------------------------------------
__global__ void concat_kernel(const float* __restrict__ x,
                              const float* __restrict__ s,
                              float* __restrict__ h) {
  int t = blockIdx.x * blockDim.x + threadIdx.x;
  if (t >= NN * 32) return;
  int    n = t >> 5, c = t & 31;
  float4 v = (c < 16) ? ((const float4*)(x + (size_t)n * DH))[c]
                      : ((const float4*)(s + (size_t)n * DH))[c - 16];
  ((float4*)(h + (size_t)n * DIN))[c] = v;
}

// GIN: hacc[col] += hsrc[row], 128 feats, 32 threads/edge x float4
__global__ void gin_scatter_kernel(const int* __restrict__ ei,
                                   const float* __restrict__ hsrc,
                                   float* __restrict__ hacc) {
  int t = blockIdx.x * blockDim.x + threadIdx.x;
  if (t >= NE * 32) return;
  int    e = t >> 5, c = t & 31;
  int    row = ei[e], col = ei[NE + e];
  float4 v   = ((const float4*)(hsrc + (size_t)row * DIN))[c];
  float* dst = hacc + (size_t)col * DIN + c * 4;
  atomicAdd(dst + 0, v.x);
  atomicAdd(dst + 1, v.y);
  atomicAdd(dst + 2, v.z);
  atomicAdd(dst + 3, v.w);
}

// GCN: sagg[col] += dinv[row]*dinv[col]*sw[row], 64 feats, 16 threads/edge
__global__ void gcn_scatter_kernel(const int* __restrict__ ei,
                                   const float* __restrict__ dinv,
                                   const float* __restrict__ sw,
                                   float* __restrict__ sagg) {
  int t = blockIdx.x * blockDim.x + threadIdx.x;
  if (t >= NE * 16) return;
  int    e = t >> 4, c = t & 15;
  int    row = ei[e], col = ei[NE + e];
  float  nrm = dinv[row] * dinv[col];
  float4 v   = ((const float4*)(sw + (size_t)row * DH))[c];
  float* dst = sagg + (size_t)col * DH + c * 4;
  atomicAdd(dst + 0, nrm * v.x);
  atomicAdd(dst + 1, nrm * v.y);
  atomicAdd(dst + 2, nrm * v.z);
  atomicAdd(dst + 3, nrm * v.w);
}

// s = tanh(sagg + dinv^2 * sw + b)
__global__ void gcn_update_kernel(const float* __restrict__ sagg,
                                  const float* __restrict__ sw,
                                  const float* __restrict__ dinv,
                                  const float* __restrict__ b,
                                  float* __restrict__ s) {
  int t = blockIdx.x * blockDim.x + threadIdx.x;
  if (t >= NN * DH) return;
  int   n  = t >> 6, f = t & 63;
  float dv = dinv[n];
  s[t]     = tanhf(sagg[t] + dv * dv * sw[t] + b[f]);
}

// global_add_pool: xp[batch[n]] += xw[n]
__global__ void pool_kernel(const float* __restrict__ xw,
                            const int* __restrict__ batch,
                            float* __restrict__ xp) {
  int t = blockIdx.x * blockDim.x + threadIdx.x;
  if (t >= NN * 16) return;
  int    n = t >> 4, c = t & 15;
  int    g = batch[n];
  float4 v = ((const float4*)(xw + (size_t)n * DH))[c];
  float* dst = xp + (size_t)g * DH + c * 4;
  atomicAdd(dst + 0, v.x);
  atomicAdd(dst + 1, v.y);
  atomicAdd(dst + 2, v.z);
  atomicAdd(dst + 3, v.w);
}

// y = log_softmax(xp @ ro_w + ro_b), one block (16 threads) per graph
__global__ void readout_lsm_kernel(const float* __restrict__ xp,
                                   const float* __restrict__ w,
                                   const float* __restrict__ b,
                                   float* __restrict__ y) {
  __shared__ float logits[DOUT];
  __shared__ float red[2];
  int   g   = blockIdx.x;
  int   o   = threadIdx.x;  // 0..15
  float acc = b[o];
  for (int k = 0; k < DH; ++k) acc += xp[g * DH + k] * w[k * DOUT + o];
  logits[o] = acc;
  __syncthreads();
  if (o == 0) {
    float mx = logits[0];
    for (int i = 1; i < DOUT; ++i) mx = fmaxf(mx, logits[i]);
    float se = 0.0f;
    for (int i = 0; i < DOUT; ++i) se += expf(logits[i] - mx);
    red[0] = mx;
    red[1] = logf(se);
  }
  __syncthreads();
  y[g * DOUT + o] = logits[o] - red[0] - red[1];
}

// ---------------------------------------------------------------------------
extern "C" void kernel_launch(void* const* d_in, const int* in_sizes, int n_in,
                              void* d_out, int out_size, void* d_ws,
                              size_t ws_size, hipStream_t stream) {
  const float* x_in   = (const float*)d_in[0];
  const float* s_in   = (const float*)d_in[1];
  const int*   ei     = (const int*)d_in[2];
  const int*   batch  = (const int*)d_in[3];
  const float* pre_w  = (const float*)d_in[4];
  const float* pre_b  = (const float*)d_in[5];
  const float* emb_w  = (const float*)d_in[6];
  const float* emb_b  = (const float*)d_in[7];
  const float* gin_w1 = (const float*)d_in[8];
  const float* gin_b1 = (const float*)d_in[9];
  const float* gin_w2 = (const float*)d_in[10];
  const float* gin_b2 = (const float*)d_in[11];
  const float* gcn_w  = (const float*)d_in[12];
  const float* gcn_b  = (const float*)d_in[13];
  const float* whp_w  = (const float*)d_in[14];
  const float* whp_b  = (const float*)d_in[15];
  const float* post_w = (const float*)d_in[16];
  const float* post_b = (const float*)d_in[17];
  const float* ro_w   = (const float*)d_in[18];
  const float* ro_b   = (const float*)d_in[19];

  // workspace layout (floats)
  float* ws   = (float*)d_ws;
  float* deg  = ws;                 // N
  float* dinv = deg + NN;           // N
  float* xbuf = dinv + NN;          // N*64
  float* sbuf = xbuf + (size_t)NN * DH;   // N*64
  float* hsrc = sbuf + (size_t)NN * DH;   // N*128
  float* hacc = hsrc + (size_t)NN * DIN;  // N*128
  float* tbuf = hacc + (size_t)NN * DIN;  // N*64 (MLP hidden / whp out)
  float* xp   = tbuf + (size_t)NN * DH;   // G*64 (pool accumulator)
  // GCN buffers alias hacc (dead after the GIN GEMM consumes it)
  float* swbuf = hacc;
  float* sagg  = hacc + (size_t)NN * DH;

  float* out_xp = (float*)d_out;            // [G,64] post output
  float* out_y  = out_xp + (size_t)NG * DH; // [G,16] log-softmax

  const int TB = 256;
  auto cdiv = [](long long a, long long b) { return (int)((a + b - 1) / b); };

  // degrees -> dinv
  hipMemsetAsync(deg, 0, (size_t)NN * sizeof(float), stream);
  edge_deg_kernel<<<cdiv(NE, TB), TB, 0, stream>>>(ei, deg);
  dinv_kernel<<<cdiv(NN, TB), TB, 0, stream>>>(deg, dinv);

  // x = x@pre_w + pre_b ; s = s@emb_w + emb_b   (WMMA GEMMs)
  gemm_f32_wmma<DIN, DH, 0, false><<<NN / 32, 256, 0, stream>>>(
      x_in, nullptr, pre_w, pre_b, xbuf);
  gemm_f32_wmma<DSE, DH, 0, false><<<NN / 32, 256, 0, stream>>>(
      s_in, nullptr, emb_w, emb_b, sbuf);

  for (int l = 0; l < NL; ++l) {
    // h = concat(x, s)
    concat_kernel<<<cdiv((long long)NN * 32, TB), TB, 0, stream>>>(xbuf, sbuf,
                                                                   hsrc);
    // agg = segment_sum(h[row], col)
    hipMemsetAsync(hacc, 0, (size_t)NN * DIN * sizeof(float), stream);
    gin_scatter_kernel<<<cdiv((long long)NE * 32, TB), TB, 0, stream>>>(
        ei, hsrc, hacc);
    // x = relu( relu((h+agg)@w1 + b1) @ w2 + b2 )   (sum fused into staging)
    gemm_f32_wmma<DIN, DH, 1, true><<<NN / 32, 256, 0, stream>>>(
        hsrc, hacc, gin_w1 + (size_t)l * DIN * DH, gin_b1 + l * DH, tbuf);
    gemm_f32_wmma<DH, DH, 1, false><<<NN / 32, 256, 0, stream>>>(
        tbuf, nullptr, gin_w2 + (size_t)l * DH * DH, gin_b2 + l * DH, xbuf);
    // GCN on s
    gemm_f32_wmma<DH, DH, 0, false><<<NN / 32, 256, 0, stream>>>(
        sbuf, nullptr, gcn_w + (size_t)l * DH * DH, nullptr, swbuf);
    hipMemsetAsync(sagg, 0, (size_t)NN * DH * sizeof(float), stream);
    gcn_scatter_kernel<<<cdiv((long long)NE * 16, TB), TB, 0, stream>>>(
        ei, dinv, swbuf, sagg);
    gcn_update_kernel<<<cdiv((long long)NN * DH, TB), TB, 0, stream>>>(
        sagg, swbuf, dinv, gcn_b + l * DH, sbuf);
  }

  // x = concat(x,s) @ whp_w + whp_b
  concat_kernel<<<cdiv((long long)NN * 32, TB), TB, 0, stream>>>(xbuf, sbuf,
                                                                 hsrc);
  gemm_f32_wmma<DIN, DH, 0, false><<<NN / 32, 256, 0, stream>>>(
      hsrc, nullptr, whp_w, whp_b, tbuf);

  // global_add_pool
  hipMemsetAsync(xp, 0, (size_t)NG * DH * sizeof(float), stream);
  pool_kernel<<<cdiv((long long)NN * 16, TB), TB, 0, stream>>>(tbuf, batch, xp);

  // xp = relu(xp @ post_w + post_b)  -> first output
  gemm_f32_wmma<DH, DH, 1, false><<<NG / 32, 256, 0, stream>>>(
      xp, nullptr, post_w, post_b, out_xp);

  // y = log_softmax(xp @ ro_w + ro_b) -> second output
  readout_lsm_kernel<<<NG, DOUT, 0, stream>>>(out_xp, ro_w, ro_b, out_y);
}